// AggrSum_38560216383546
// MI455X (gfx1250) — compile-verified
//
#include <hip/hip_runtime.h>
#include <stdint.h>

// Segment-sum (scatter-add) of H[N,128] f32 into out[V,128] by index X_node.
// Memory-bound: ~851 MB traffic -> ~36us floor at 23.3 TB/s.
// Strategy: async-copy H tiles to LDS (ASYNCcnt double-buffer, b128 per lane),
// read back with ds_load_b128, then fire-and-forget global_atomic_add_f32 at
// DEV scope (output is L2-resident: 25.6 MB << 192 MB L2).

typedef float float4v __attribute__((ext_vector_type(4)));
typedef int   int4v   __attribute__((ext_vector_type(4)));
typedef __attribute__((address_space(1))) int4v*   g_int4p;  // global int4*
typedef __attribute__((address_space(3))) int4v*   l_int4p;  // LDS int4*
typedef __attribute__((address_space(3))) float4v* l_f4p;    // LDS float4*

#define BLOCK 256
#define WAVES_PER_BLOCK (BLOCK / 32)
#define CHUNK 4            // rows per wave per LDS buffer (4 rows * 512B = 2KB)
#define D 128              // feature dim (floats per row)
#define ROW_BYTES (D * 4)  // 512
#define GRID_MAIN 2048

// ---------------------------------------------------------------- async copy
__device__ __forceinline__ void async_copy_b128(const char* Hc, uint32_t byteoff,
                                                l_f4p lds_dst) {
  const float* src = (const float*)(Hc + byteoff);
#if __has_builtin(__builtin_amdgcn_global_load_async_to_lds_b128)
  __builtin_amdgcn_global_load_async_to_lds_b128(
      (g_int4p)(void*)src, (l_int4p)lds_dst, /*imm offset*/ 0, /*cpol*/ 0);
#else
  uint32_t lds_off = (uint32_t)(uintptr_t)lds_dst;
  asm volatile("global_load_async_to_lds_b128 %0, %1, off"
               :: "v"(lds_off), "v"((uint64_t)(uintptr_t)src)
               : "memory");
#endif
}

__device__ __forceinline__ void wait_async_le4() {
#if __has_builtin(__builtin_amdgcn_s_wait_asynccnt)
  __builtin_amdgcn_s_wait_asynccnt(4);
#else
  asm volatile("s_wait_asynccnt 4" ::: "memory");
#endif
}

__device__ __forceinline__ void wait_async_0() {
#if __has_builtin(__builtin_amdgcn_s_wait_asynccnt)
  __builtin_amdgcn_s_wait_asynccnt(0);
#else
  asm volatile("s_wait_asynccnt 0" ::: "memory");
#endif
}

// ---------------------------------------- no-return dev-scope fadd x4 (SADDR)
__device__ __forceinline__ void atomic_add4_f32(const float* out_base,
                                                uint32_t byteoff, float4v d) {
  asm volatile(
      "global_atomic_add_f32 %1, %2, %0 scope:SCOPE_DEV\n\t"
      "global_atomic_add_f32 %1, %3, %0 offset:4 scope:SCOPE_DEV\n\t"
      "global_atomic_add_f32 %1, %4, %0 offset:8 scope:SCOPE_DEV\n\t"
      "global_atomic_add_f32 %1, %5, %0 offset:12 scope:SCOPE_DEV"
      :: "s"((uint64_t)(uintptr_t)out_base), "v"(byteoff),
         "v"(d.x), "v"(d.y), "v"(d.z), "v"(d.w)
      : "memory");
}

// ------------------------------------------------------------ scatter kernel
__global__ __launch_bounds__(BLOCK) void AggrSum_scatter(
    const float* __restrict__ H, const int* __restrict__ Xn,
    float* __restrict__ out, int N) {
  // per-wave double buffer: 2 * CHUNK rows * 512B = 4KB/wave, 32KB/block
  __shared__ __align__(16) float lds[WAVES_PER_BLOCK * 2 * CHUNK * D];

  const uint32_t lane = threadIdx.x & 31u;
  const uint32_t w = threadIdx.x >> 5;
  // as(3) pointers in float4 units; each lane owns one float4 column per row
  l_f4p wavebase = (l_f4p)(void*)&lds[w * 2 * CHUNK * D];
  l_f4p buf0 = wavebase + lane;
  l_f4p buf1 = wavebase + (CHUNK * (D / 4)) + lane;

  const char* Hc = (const char*)H;
  const int nchunks = (N + CHUNK - 1) / CHUNK;
  const int stride = (int)(gridDim.x * WAVES_PER_BLOCK);
  int chunk = (int)(blockIdx.x * WAVES_PER_BLOCK + w);
  if (chunk >= nchunks) return;

  // prologue: issue async loads for first chunk into buf0
  {
    uint32_t row0 = (uint32_t)chunk * CHUNK;
    uint32_t base_off = row0 * ROW_BYTES + lane * 16u;
    if (N - (int)row0 >= CHUNK) {
#pragma unroll
      for (int j = 0; j < CHUNK; ++j)
        async_copy_b128(Hc, base_off + (uint32_t)j * ROW_BYTES,
                        buf0 + j * (D / 4));
    } else {
      int rows = N - (int)row0;
      for (int j = 0; j < rows; ++j)
        async_copy_b128(Hc, base_off + (uint32_t)j * ROW_BYTES,
                        buf0 + j * (D / 4));
    }
  }

  int cur = 0;
  for (;;) {
    // issue async loads for next chunk into the other buffer
    int next = chunk + stride;
    int nrows = 0;
    if (next < nchunks) {
      uint32_t nr0 = (uint32_t)next * CHUNK;
      l_f4p nb = cur ? buf0 : buf1;
      uint32_t base_off = nr0 * ROW_BYTES + lane * 16u;
      if (N - (int)nr0 >= CHUNK) {
        nrows = CHUNK;
#pragma unroll
        for (int j = 0; j < CHUNK; ++j)
          async_copy_b128(Hc, base_off + (uint32_t)j * ROW_BYTES,
                          nb + j * (D / 4));
      } else {
        nrows = N - (int)nr0;
        for (int j = 0; j < nrows; ++j)
          async_copy_b128(Hc, base_off + (uint32_t)j * ROW_BYTES,
                          nb + j * (D / 4));
      }
    }

    // wait for current buffer (async loads complete in order):
    // outstanding after this wait <= nrows (the next chunk's loads)
    if (nrows == CHUNK) wait_async_le4(); else wait_async_0();

    // scatter-add current buffer
    uint32_t row0 = (uint32_t)chunk * CHUNK;
    l_f4p cb = cur ? buf1 : buf0;
    const uint32_t lane_off = lane * 16u;
    if (N - (int)row0 >= CHUNK) {
      // one b128 load of 4 indices (row0 % 4 == 0 -> 16B aligned)
      int4v idx4 = *(const int4v*)(Xn + row0);
      float4v d0 = cb[0 * (D / 4)];
      float4v d1 = cb[1 * (D / 4)];
      float4v d2 = cb[2 * (D / 4)];
      float4v d3 = cb[3 * (D / 4)];
      atomic_add4_f32(out, (uint32_t)idx4.x * ROW_BYTES + lane_off, d0);
      atomic_add4_f32(out, (uint32_t)idx4.y * ROW_BYTES + lane_off, d1);
      atomic_add4_f32(out, (uint32_t)idx4.z * ROW_BYTES + lane_off, d2);
      atomic_add4_f32(out, (uint32_t)idx4.w * ROW_BYTES + lane_off, d3);
    } else {
      int rows = N - (int)row0;
      for (int j = 0; j < rows; ++j) {
        float4v d = cb[j * (D / 4)];
        uint32_t v = (uint32_t)Xn[row0 + (uint32_t)j];
        atomic_add4_f32(out, v * ROW_BYTES + lane_off, d);
      }
    }

    if (next >= nchunks) break;
    chunk = next;
    cur ^= 1;
  }
}

// --------------------------------------------------------------- zero kernel
__global__ __launch_bounds__(BLOCK) void AggrSum_zero(float* __restrict__ out,
                                                      int n) {
  int i = (int)(blockIdx.x * BLOCK + threadIdx.x) * 4;
  if (i + 3 < n) {
    *(float4v*)(out + i) = (float4v){0.f, 0.f, 0.f, 0.f};
  } else {
    for (int k = i; k < n; ++k) out[k] = 0.f;
  }
}

// ------------------------------------------------------------------- launch
extern "C" void kernel_launch(void* const* d_in, const int* in_sizes, int n_in,
                              void* d_out, int out_size, void* d_ws, size_t ws_size,
                              hipStream_t stream) {
  const float* H = (const float*)d_in[0];
  const int* Xn = (const int*)d_in[1];
  float* out = (float*)d_out;
  const int N = in_sizes[1];                 // number of rows

  // 1) zero the accumulator (harness poisons d_out; atomics accumulate)
  {
    int n4 = (out_size + 3) / 4;
    int grid = (n4 + BLOCK - 1) / BLOCK;
    AggrSum_zero<<<grid, BLOCK, 0, stream>>>(out, out_size);
  }

  // 2) streaming scatter-add
  {
    int nchunks = (N + CHUNK - 1) / CHUNK;
    int maxg = (nchunks + WAVES_PER_BLOCK - 1) / WAVES_PER_BLOCK;
    int grid = maxg < GRID_MAIN ? maxg : GRID_MAIN;
    if (grid < 1) grid = 1;
    AggrSum_scatter<<<grid, BLOCK, 0, stream>>>(H, Xn, out, N);
  }
}